// GCN_64750926954705
// MI455X (gfx1250) — compile-verified
//
#include <hip/hip_runtime.h>

typedef __attribute__((ext_vector_type(2))) float v2f;
typedef __attribute__((ext_vector_type(8))) float v8f;

#define D_IN  128
#define D_HID 64
#define D_OUT 32

// LDS strides for transposed weight tiles: +2 floats keeps 8B alignment for
// v2f (k even) loads and places consecutive columns 2 banks apart -> the 16
// columns of a half-wave hit 32 distinct banks (conflict-free b64 reads).
#define SW1_STRIDE (D_IN + 2)   // 130
#define SW2_STRIDE (D_HID + 2)  // 66

// ---------------- degree / normalization ----------------

__global__ __launch_bounds__(256) void k_init_deg(float* deg, int n) {
  int i = blockIdx.x * 256 + threadIdx.x;
  if (i < n) deg[i] = 1.0f;  // self-loop contributes 1 to every node's degree
}

__global__ __launch_bounds__(256) void k_count_deg(const int* __restrict__ ei,
                                                   float* deg, int E) {
  int e = blockIdx.x * 256 + threadIdx.x;
  if (e < E) atomicAdd(&deg[ei[E + e]], 1.0f);  // dst row of edge_index
}

__global__ __launch_bounds__(256) void k_dinv(float* deg, int n) {
  int i = blockIdx.x * 256 + threadIdx.x;
  if (i < n) deg[i] = rsqrtf(deg[i]);  // deg >= 1 always (self-loops)
}

// ---------------- GEMM1: xw1 = x @ W1  (fp32 WMMA 16x16x4) ----------------
// Block = 256 threads = 8 waves. Wave w computes rows [blk*128 + w*16, +16),
// all 64 output columns as 4 16x16 accumulator tiles. W1 transposed in LDS.

__global__ __launch_bounds__(256) void k_gemm1(const float* __restrict__ x,
                                               const float* __restrict__ W1,
                                               float* __restrict__ xw1, int n) {
  __shared__ float sW[D_HID * SW1_STRIDE];  // ~33.3 KB, sW[col][k]
  for (int i = threadIdx.x; i < D_IN * D_HID; i += 256) {
    int k = i >> 6;   // W1 row   (coalesced global read)
    int c = i & 63;   // W1 col   (consecutive lanes -> 2-bank stride, no conflict)
    sW[c * SW1_STRIDE + k] = W1[i];
  }
  __syncthreads();

  const int wave  = threadIdx.x >> 5;
  const int lane  = threadIdx.x & 31;
  const int mLane = lane & 15;        // row (A) / col (B,D) within tile
  const int kHalf = (lane >> 4) * 2;  // lanes 0-15: K=0,1 ; lanes 16-31: K=2,3
  const int rowBase = (blockIdx.x * 8 + wave) * 16;

  v8f acc[4];
  const v8f zero = {0, 0, 0, 0, 0, 0, 0, 0};
#pragma unroll
  for (int t = 0; t < 4; ++t) acc[t] = zero;

  int aRow = rowBase + mLane;
  if (aRow >= n) aRow = n - 1;  // clamp loads; EXEC stays all-1 for WMMA
  const float* aPtr = x + (size_t)aRow * D_IN + kHalf;

  for (int k = 0; k < D_IN; k += 4) {
    const v2f a = *(const v2f*)(aPtr + k);  // single b64 load
#pragma unroll
    for (int t = 0; t < 4; ++t) {
      const int col = t * 16 + mLane;
      const v2f b = *(const v2f*)&sW[col * SW1_STRIDE + k + kHalf];  // ds_load_b64
      acc[t] = __builtin_amdgcn_wmma_f32_16x16x4_f32(
          false, a, false, b, (short)0, acc[t], false, false);
    }
  }

  // D layout: VGPR r -> lanes 0-15: M=r, N=lane ; lanes 16-31: M=r+8, N=lane-16
  const int mOff = (lane >> 4) * 8;
  const int row0 = rowBase + mOff;
  if (rowBase + 16 <= n) {  // wave-uniform fast path: unguarded stores
#pragma unroll
    for (int t = 0; t < 4; ++t) {
      float* p = xw1 + (size_t)row0 * D_HID + t * 16 + mLane;
#pragma unroll
      for (int r = 0; r < 8; ++r) p[r * D_HID] = acc[t][r];
    }
  } else {
#pragma unroll
    for (int t = 0; t < 4; ++t) {
      float* p = xw1 + (size_t)row0 * D_HID + t * 16 + mLane;
#pragma unroll
      for (int r = 0; r < 8; ++r)
        if (row0 + r < n) p[r * D_HID] = acc[t][r];
    }
  }
}

// ---------------- layer-1 aggregation ----------------

__global__ __launch_bounds__(256) void k_selfloop1(const float* __restrict__ xw1,
                                                   const float* __restrict__ dinv,
                                                   float* __restrict__ h, int n) {
  int tid = blockIdx.x * 256 + threadIdx.x;  // one float4 per thread, 16 per row
  int i = tid >> 4, f4 = tid & 15;
  if (i >= n) return;
  float s = dinv[i] * dinv[i];
  float4 v = ((const float4*)(xw1 + (size_t)i * D_HID))[f4];
  float4 o = {v.x * s, v.y * s, v.z * s, v.w * s};
  ((float4*)(h + (size_t)i * D_HID))[f4] = o;
}

__global__ __launch_bounds__(256) void k_agg1(const int* __restrict__ ei,
                                              const float* __restrict__ dinv,
                                              const float* __restrict__ xw1,
                                              float* __restrict__ h, int E) {
  int tid = blockIdx.x * 256 + threadIdx.x;  // 16 lanes per edge
  int e = tid >> 4, f4 = tid & 15;
  if (e >= E) return;
  int src = ei[e], dst = ei[E + e];
  float norm = dinv[src] * dinv[dst];
  float4 v = ((const float4*)(xw1 + (size_t)src * D_HID))[f4];
  float* o = h + (size_t)dst * D_HID + f4 * 4;
  atomicAdd(o + 0, v.x * norm);
  atomicAdd(o + 1, v.y * norm);
  atomicAdd(o + 2, v.z * norm);
  atomicAdd(o + 3, v.w * norm);
}

// ---------------- GEMM2: hw2 = relu(h + b1) @ W2  (bias+relu fused into A load) ----

__global__ __launch_bounds__(256) void k_gemm2(const float* __restrict__ h,
                                               const float* __restrict__ b1,
                                               const float* __restrict__ W2,
                                               float* __restrict__ hw2, int n) {
  __shared__ float sW[D_OUT * SW2_STRIDE];  // sW[col][k], ~8.3 KB
  __shared__ float sb[D_HID];
  for (int i = threadIdx.x; i < D_HID * D_OUT; i += 256) {
    int k = i >> 5;   // W2 row
    int c = i & 31;   // W2 col
    sW[c * SW2_STRIDE + k] = W2[i];
  }
  if (threadIdx.x < D_HID) sb[threadIdx.x] = b1[threadIdx.x];
  __syncthreads();

  const int wave  = threadIdx.x >> 5;
  const int lane  = threadIdx.x & 31;
  const int mLane = lane & 15;
  const int kHalf = (lane >> 4) * 2;
  const int rowBase = (blockIdx.x * 8 + wave) * 16;

  v8f acc[2];
  const v8f zero = {0, 0, 0, 0, 0, 0, 0, 0};
  acc[0] = zero;
  acc[1] = zero;

  int aRow = rowBase + mLane;
  if (aRow >= n) aRow = n - 1;
  const float* aPtr = h + (size_t)aRow * D_HID + kHalf;

  for (int k = 0; k < D_HID; k += 4) {
    const v2f araw = *(const v2f*)(aPtr + k);  // single b64 load
    v2f a;
    a.x = fmaxf(araw.x + sb[k + kHalf], 0.0f);
    a.y = fmaxf(araw.y + sb[k + kHalf + 1], 0.0f);
#pragma unroll
    for (int t = 0; t < 2; ++t) {
      const int col = t * 16 + mLane;
      const v2f b = *(const v2f*)&sW[col * SW2_STRIDE + k + kHalf];
      acc[t] = __builtin_amdgcn_wmma_f32_16x16x4_f32(
          false, a, false, b, (short)0, acc[t], false, false);
    }
  }

  const int mOff = (lane >> 4) * 8;
  const int row0 = rowBase + mOff;
  if (rowBase + 16 <= n) {  // wave-uniform fast path
#pragma unroll
    for (int t = 0; t < 2; ++t) {
      float* p = hw2 + (size_t)row0 * D_OUT + t * 16 + mLane;
#pragma unroll
      for (int r = 0; r < 8; ++r) p[r * D_OUT] = acc[t][r];
    }
  } else {
#pragma unroll
    for (int t = 0; t < 2; ++t) {
      float* p = hw2 + (size_t)row0 * D_OUT + t * 16 + mLane;
#pragma unroll
      for (int r = 0; r < 8; ++r)
        if (row0 + r < n) p[r * D_OUT] = acc[t][r];
    }
  }
}

// ---------------- layer-2 aggregation (into d_out, bias folded into init) ----

__global__ __launch_bounds__(256) void k_initout(const float* __restrict__ hw2,
                                                 const float* __restrict__ dinv,
                                                 const float* __restrict__ b2,
                                                 float* __restrict__ out, int n) {
  int tid = blockIdx.x * 256 + threadIdx.x;  // 8 float4 per row
  int i = tid >> 3, f4 = tid & 7;
  if (i >= n) return;
  float s = dinv[i] * dinv[i];
  float4 v = ((const float4*)(hw2 + (size_t)i * D_OUT))[f4];
  const float4 bb = ((const float4*)b2)[f4];
  float4 o = {v.x * s + bb.x, v.y * s + bb.y, v.z * s + bb.z, v.w * s + bb.w};
  ((float4*)(out + (size_t)i * D_OUT))[f4] = o;
}

__global__ __launch_bounds__(256) void k_agg2(const int* __restrict__ ei,
                                              const float* __restrict__ dinv,
                                              const float* __restrict__ hw2,
                                              float* __restrict__ out, int E) {
  int tid = blockIdx.x * 256 + threadIdx.x;  // 8 lanes per edge
  int e = tid >> 3, f4 = tid & 7;
  if (e >= E) return;
  int src = ei[e], dst = ei[E + e];
  float norm = dinv[src] * dinv[dst];
  float4 v = ((const float4*)(hw2 + (size_t)src * D_OUT))[f4];
  float* o = out + (size_t)dst * D_OUT + f4 * 4;
  atomicAdd(o + 0, v.x * norm);
  atomicAdd(o + 1, v.y * norm);
  atomicAdd(o + 2, v.z * norm);
  atomicAdd(o + 3, v.w * norm);
}

// ---------------- launcher ----------------

extern "C" void kernel_launch(void* const* d_in, const int* in_sizes, int n_in,
                              void* d_out, int out_size, void* d_ws, size_t ws_size,
                              hipStream_t stream) {
  const float* x  = (const float*)d_in[0];
  const int*   ei = (const int*)d_in[1];
  const float* W1 = (const float*)d_in[2];
  const float* b1 = (const float*)d_in[3];
  const float* W2 = (const float*)d_in[4];
  const float* b2 = (const float*)d_in[5];
  float* out = (float*)d_out;

  const int N = in_sizes[0] / D_IN;
  const int E = in_sizes[1] / 2;

  float* deg  = (float*)d_ws;               // N        (becomes dinv in place)
  float* xw1  = deg + N;                    // N * 64
  float* hraw = xw1 + (size_t)N * D_HID;    // N * 64
  float* hw2  = hraw + (size_t)N * D_HID;   // N * 32
  // total: N*161*4 B ~= 32.2 MB of workspace

  const int nbN = (N + 255) / 256;
  k_init_deg<<<nbN, 256, 0, stream>>>(deg, N);
  k_count_deg<<<(E + 255) / 256, 256, 0, stream>>>(ei, deg, E);
  k_dinv<<<nbN, 256, 0, stream>>>(deg, N);

  k_gemm1<<<(N + 127) / 128, 256, 0, stream>>>(x, W1, xw1, N);
  k_selfloop1<<<(N * 16 + 255) / 256, 256, 0, stream>>>(xw1, deg, hraw, N);
  k_agg1<<<(E * 16 + 255) / 256, 256, 0, stream>>>(ei, deg, xw1, hraw, E);

  k_gemm2<<<(N + 127) / 128, 256, 0, stream>>>(hraw, b1, W2, hw2, N);
  k_initout<<<(N * 8 + 255) / 256, 256, 0, stream>>>(hw2, deg, b2, out, N);
  k_agg2<<<(E * 8 + 255) / 256, 256, 0, stream>>>(ei, deg, hw2, out, E);
}